// LGCN_32109175504989
// MI455X (gfx1250) — compile-verified
//
#include <hip/hip_runtime.h>

typedef __attribute__((ext_vector_type(2))) float v2f;
typedef __attribute__((ext_vector_type(8))) float v8f;

#define N_NODES 1600
#define HDIM    48
#define MAXNBR  192

__device__ inline v8f wmma_f32(v2f a, v2f b, v8f c) {
  // D = A(16x4, f32) * B(4x16, f32) + C(16x16, f32)
  return __builtin_amdgcn_wmma_f32_16x16x4_f32(
      /*neg_a=*/false, a, /*neg_b=*/false, b,
      /*c_mod=*/(short)0, c, /*reuse_a=*/false, /*reuse_b=*/false);
}

// Compile-time-specialized fp32 WMMA GEMM.
// C[row0:row0+16, col0:col0+16*NT] = A * B
// A row-major [M x K] (LDA), B row-major [K x BCOLS] (LDB), C (LDC).
// Ragged BCOLS/CCOLS handled with clamped addresses + v_cndmask selects
// (no exec-mask predication); when BCOLS/CCOLS are multiples of 16 the
// checks fold away at compile time. One wave per 16x(16*NT) C strip; the
// A fragment is shared across the NT accumulators.
template <int K, int LDA, int LDB, int LDC, int BCOLS, int CCOLS, int NT>
__global__ __launch_bounds__(32)
void gemm_tile(const float* __restrict__ A, const float* __restrict__ B,
               float* __restrict__ C) {
  const int lane = threadIdx.x & 31;
  const int row0 = blockIdx.x * 16;
  const int col0 = blockIdx.y * (16 * NT);
  const int m    = lane & 15;          // A row / B col / C col this lane serves
  const int kb   = (lane >> 4) << 1;   // K sub-offset: 0 (lanes 0-15) or 2 (16-31)

  const float* __restrict__ arowb = A + (size_t)(row0 + m) * LDA;  // un-offset base
  const float* __restrict__ arow  = arowb + kb;

  v8f acc[NT] = {};

  int  cb[NT];
  bool bv[NT];
  #pragma unroll
  for (int t = 0; t < NT; ++t) {
    int c = col0 + t * 16 + m;
    bv[t] = ((BCOLS & 15) == 0) || (c < BCOLS);  // compile-time true for full tiles
    cb[t] = bv[t] ? c : 0;                       // clamped -> address always valid
  }

  constexpr int K4 = K & ~3;
  #pragma unroll 4
  for (int k = 0; k < K4; k += 4) {
    v2f a;
    a.x = arow[k];
    a.y = arow[k + 1];
    #pragma unroll
    for (int t = 0; t < NT; ++t) {
      float b0 = B[(size_t)(k + kb)     * LDB + cb[t]];
      float b1 = B[(size_t)(k + kb + 1) * LDB + cb[t]];
      v2f b;
      b.x = bv[t] ? b0 : 0.0f;
      b.y = bv[t] ? b1 : 0.0f;
      acc[t] = wmma_f32(a, b, acc[t]);
    }
  }
  if constexpr (K4 < K) {              // ragged-K tail, clamped addresses
    const int  k0 = K4 + kb;
    const bool v0 = (k0 < K), v1 = (k0 + 1 < K);
    const int  i0 = v0 ? k0 : 0;
    const int  i1 = v1 ? k0 + 1 : 0;
    float a0 = arowb[i0], a1 = arowb[i1];
    v2f a;
    a.x = v0 ? a0 : 0.0f;
    a.y = v1 ? a1 : 0.0f;
    #pragma unroll
    for (int t = 0; t < NT; ++t) {
      float b0 = B[(size_t)i0 * LDB + cb[t]];
      float b1 = B[(size_t)i1 * LDB + cb[t]];
      v2f b;
      b.x = (v0 && bv[t]) ? b0 : 0.0f;
      b.y = (v1 && bv[t]) ? b1 : 0.0f;
      acc[t] = wmma_f32(a, b, acc[t]);
    }
  }

  const int half = (lane >> 4) * 8;    // C/D layout: vgpr r -> row half+r, col = lane&15
  #pragma unroll
  for (int t = 0; t < NT; ++t) {
    int  cc = col0 + t * 16 + m;
    bool cv = ((CCOLS & 15) == 0) || (cc < CCOLS);
    if (cv) {
      #pragma unroll
      for (int r = 0; r < 8; ++r)
        C[(size_t)(row0 + half + r) * LDC + cc] = acc[t][r];
    }
  }
}

// Final stage: out = mask ? (adj @ hw)[:, :7] : 0   (hw padded to 16 cols, zeros in 7..15)
__global__ __launch_bounds__(32)
void gemm_adj_out(const float* __restrict__ adj, const float* __restrict__ hw,
                  const int* __restrict__ mask, float* __restrict__ out) {
  const int lane = threadIdx.x & 31;
  const int row0 = blockIdx.x * 16;
  const int m    = lane & 15;
  const int kb   = (lane >> 4) << 1;
  const float* __restrict__ arow = adj + (size_t)(row0 + m) * N_NODES + kb;
  const float* __restrict__ bcol = hw + m;   // ldb = 16
  v8f acc = {};
  #pragma unroll 4
  for (int k = 0; k < N_NODES; k += 4) {
    v2f a, b;
    a.x = arow[k];
    a.y = arow[k + 1];
    b.x = bcol[(size_t)(k + kb)     * 16];
    b.y = bcol[(size_t)(k + kb + 1) * 16];
    acc = wmma_f32(a, b, acc);
  }
  const int half = (lane >> 4) * 8;
  if (m < 7) {
    #pragma unroll
    for (int r = 0; r < 8; ++r) {
      int rr = row0 + half + r;
      out[(size_t)rr * 7 + m] = mask[rr] ? acc[r] : 0.0f;
    }
  }
}

// One wave per node: sparse neighbor scan -> per-channel top-8 (zero-floored,
// exactly matching top_k over {neighbor values} U {many zeros}) -> conv 9->5 ->
// conv 5->1 -> batchnorm -> append 8 channels into H at column `outcol`.
__global__ __launch_bounds__(32)
void lgconv(const float* __restrict__ adj, float* __restrict__ H,
            const float* __restrict__ Wa,  // [5, F, mid]
            const float* __restrict__ Wb,  // [5, mid, 8]
            const float* __restrict__ bng, const float* __restrict__ bnb,
            const float* __restrict__ bnm, const float* __restrict__ bnv,
            int F, int mid, int outcol) {
  __shared__ int   nbr[MAXNBR];
  __shared__ int   cnt;
  __shared__ float t[9][HDIM];     // rows: [own, top1..top8], cols: channels
  __shared__ float cbuf[5][24];    // conv1 output, mid <= 24

  const int node = blockIdx.x;
  const int lane = threadIdx.x;
  if (lane == 0) cnt = 0;
  __syncthreads();

  // 1) build neighbor index list (adj is exactly 0.0/1.0)
  const float* __restrict__ arow = adj + (size_t)node * N_NODES;
  for (int j = lane; j < N_NODES; j += 32) {
    if (arow[j] != 0.0f) {
      int idx = atomicAdd(&cnt, 1);
      if (idx < MAXNBR) nbr[idx] = j;
    }
  }
  __syncthreads();
  int d = cnt; if (d > MAXNBR) d = MAXNBR;

  // 2) per-channel top-8 (init 0 == zero entries of the masked row)
  for (int f = lane; f < F; f += 32) {
    float tk[8];
    #pragma unroll
    for (int q = 0; q < 8; ++q) tk[q] = 0.0f;
    for (int nn = 0; nn < d; ++nn) {
      float v = H[(size_t)nbr[nn] * HDIM + f];
      if (v > tk[7]) {
        tk[7] = v;
        #pragma unroll
        for (int q = 7; q > 0; --q)
          if (tk[q] > tk[q - 1]) { float tmp = tk[q - 1]; tk[q - 1] = tk[q]; tk[q] = tmp; }
      }
    }
    t[0][f] = H[(size_t)node * HDIM + f];
    #pragma unroll
    for (int q = 0; q < 8; ++q) t[1 + q][f] = tk[q];
  }
  __syncthreads();

  // 3) conv1d VALID: [9,F] * [5,F,mid] -> [5,mid]
  for (int idx = lane; idx < 5 * mid; idx += 32) {
    int p = idx / mid, co = idx % mid;
    float s = 0.0f;
    for (int kw = 0; kw < 5; ++kw)
      for (int ci = 0; ci < F; ++ci)
        s += t[p + kw][ci] * Wa[(size_t)(kw * F + ci) * mid + co];
    cbuf[p][co] = s;
  }
  __syncthreads();

  // 4) conv1d VALID: [5,mid] * [5,mid,8] -> [8], then batchnorm, append to H
  if (lane < 8) {
    float s = 0.0f;
    for (int kw = 0; kw < 5; ++kw)
      for (int ci = 0; ci < mid; ++ci)
        s += cbuf[kw][ci] * Wb[(size_t)(kw * mid + ci) * 8 + lane];
    float y = bng[lane] * (s - bnm[lane]) * __frsqrt_rn(bnv[lane] + 1e-3f) + bnb[lane];
    H[(size_t)node * HDIM + outcol + lane] = y;
  }
}

extern "C" void kernel_launch(void* const* d_in, const int* in_sizes, int n_in,
                              void* d_out, int out_size, void* d_ws, size_t ws_size,
                              hipStream_t stream) {
  const float* x    = (const float*)d_in[0];
  const float* adj  = (const float*)d_in[1];
  const float* W0   = (const float*)d_in[2];
  const float* W1a  = (const float*)d_in[3];
  const float* W1b  = (const float*)d_in[4];
  const float* W2a  = (const float*)d_in[5];
  const float* W2b  = (const float*)d_in[6];
  const float* Wout = (const float*)d_in[7];
  const float* g1 = (const float*)d_in[8];
  const float* b1 = (const float*)d_in[9];
  const float* m1 = (const float*)d_in[10];
  const float* v1 = (const float*)d_in[11];
  const float* g2 = (const float*)d_in[12];
  const float* b2 = (const float*)d_in[13];
  const float* m2 = (const float*)d_in[14];
  const float* v2 = (const float*)d_in[15];
  const int*  mask = (const int*)d_in[16];
  float* out = (float*)d_out;

  // workspace layout (floats): xw[1600*32] | H[1600*48] | hw[1600*16]
  float* xw = (float*)d_ws;
  float* H  = xw + N_NODES * 32;
  float* hw = H  + N_NODES * HDIM;

  dim3 blk(32);
  const int MT = N_NODES / 16;  // 100 row tiles

  // xw = x @ W0                  [1600,1433] x [1433,32]; 2 N-tiles per wave
  gemm_tile<1433, 1433, 32, 32, 32, 32, 2><<<dim3(MT, 1), blk, 0, stream>>>(x, W0, xw);
  // H[:, 0:32] = adj @ xw        [1600,1600] x [1600,32]; 2 N-tiles per wave
  gemm_tile<1600, 1600, 32, 48, 32, 32, 2><<<dim3(MT, 1), blk, 0, stream>>>(adj, xw, H);
  // LGConv block 1: F=32 -> mid=20 -> 8, write H[:, 32:40]
  lgconv<<<dim3(N_NODES), blk, 0, stream>>>(adj, H, W1a, W1b, g1, b1, m1, v1, 32, 20, 32);
  // LGConv block 2: F=40 -> mid=24 -> 8, write H[:, 40:48]
  lgconv<<<dim3(N_NODES), blk, 0, stream>>>(adj, H, W2a, W2b, g2, b2, m2, v2, 40, 24, 40);
  // hw = H @ Wout, padded to 16 cols (cols 7..15 exactly 0 via zero-selected B)
  gemm_tile<HDIM, HDIM, 7, 16, 7, 16, 1><<<dim3(MT, 1), blk, 0, stream>>>(H, Wout, hw);
  // out = mask ? (adj @ hw)[:, :7] : 0
  gemm_adj_out<<<dim3(MT), blk, 0, stream>>>(adj, hw, mask, out);
}